// BahdanauAttention_41747082117247
// MI455X (gfx1250) — compile-verified
//
#include <hip/hip_runtime.h>
#include <hip/hip_bf16.h>
#include <math.h>

// Problem constants (from reference)
#define NB   32
#define SKL  8192
#define DIM  128

typedef __attribute__((ext_vector_type(2))) float v2f;
typedef __attribute__((ext_vector_type(8))) float v8f;

// ---------------------------------------------------------------------------
// Kernel 1: r[b][e] = sum_d q[b][d] * W2[d][e]    (tiny: 32x128)
// ---------------------------------------------------------------------------
__global__ __launch_bounds__(128) void qproj_kernel(
    const float* __restrict__ q,   // (B,1,DIM)
    const float* __restrict__ W2,  // (DIM,DIM)
    float* __restrict__ r)         // (B,DIM)
{
    const int b = blockIdx.x;
    const int j = threadIdx.x;
    float acc = 0.f;
    #pragma unroll 8
    for (int d = 0; d < DIM; ++d)
        acc += q[b * DIM + d] * W2[d * DIM + j];
    r[b * DIM + j] = acc;
}

// ---------------------------------------------------------------------------
// Kernel 2: logits[b][s] = tanh(k[b,s,:] @ W1 + r[b,:]) @ Wf + bf
// WMMA f32 16x16x4. Block = 8 waves x 16 rows = 128 rows of one batch.
// W1 staged TRANSPOSED in LDS with an XOR bank swizzle so each B fragment
// {W1[k][c], W1[k+1][c]} is one contiguous ds_load_b64 (no register fix-ups,
// pipelined dscnt waits). Two column tiles per pass with independent
// accumulators break the WMMA->WMMA RAW chain (2x ILP on the matrix pipe).
// A-panel (16x128 of k) held in 32 v2f fragments in registers.
// ---------------------------------------------------------------------------
__global__ __launch_bounds__(256) void logits_kernel(
    const float* __restrict__ kmat,  // (B,SK,DIM)
    const float* __restrict__ W1,    // (DIM,DIM)
    const float* __restrict__ Wf,    // (DIM,1)
    const float* __restrict__ bfp,   // (1,)
    const float* __restrict__ r,     // (B,DIM)
    float* __restrict__ logits)      // (B,SK)
{
    // Transposed + swizzled: sW1T[col*128 + (kidx ^ ((col&7)<<3))] = W1[kidx][col]
    __shared__ float sW1T[DIM * DIM];  // 64 KB

    const int b    = blockIdx.y;
    const int row0 = blockIdx.x * 128;
    const int tid  = threadIdx.x;

    // Fill: consecutive lanes -> consecutive kidx -> conflict-free LDS writes.
    for (int i = tid; i < DIM * DIM; i += 256) {
        const int kidx = i & 127;
        const int col  = i >> 7;
        sW1T[col * DIM + (kidx ^ ((col & 7) << 3))] = W1[kidx * DIM + col];
    }
    __syncthreads();

    const int wave    = tid >> 5;
    const int lane    = tid & 31;
    const int halfsel = lane >> 4;   // 0: K pair {0,1}; 1: K pair {2,3}
    const int lh      = lane & 15;
    const int s0      = row0 + wave * 16;
    const int sw      = (lh & 7) << 3;  // (col&7)<<3, since col = n*16+lh

    // A fragments: lane lh (half h) holds k[b][s0+lh][4kk + 2h .. +1]
    const float* krow = kmat + ((size_t)b * SKL + (size_t)(s0 + lh)) * DIM;
    v2f afrag[32];
    #pragma unroll
    for (int kk = 0; kk < 32; ++kk) {
        const int col = kk * 4 + halfsel * 2;
        afrag[kk] = *(const v2f*)(krow + col);   // 8B aligned
    }

    float logacc[8];
    #pragma unroll
    for (int i = 0; i < 8; ++i) logacc[i] = 0.f;

    #pragma unroll 1
    for (int n = 0; n < 8; n += 2) {            // 2 column tiles of 16 per pass
        v8f acc0 = {};
        v8f acc1 = {};
        const int c0 = n * 16 + lh;             // column tile n
        const int c1 = c0 + 16;                 // column tile n+1
        const float* base0 = &sW1T[c0 * DIM];
        const float* base1 = &sW1T[c1 * DIM];
        #pragma unroll
        for (int kk = 0; kk < 32; ++kk) {       // K = 128 in steps of 4
            const int k0 = kk * 4 + halfsel * 2;
            const int ks = k0 ^ sw;             // swizzled, still even -> 8B aligned
            const v2f bf0 = *(const v2f*)(base0 + ks);
            const v2f bf1 = *(const v2f*)(base1 + ks);
            acc0 = __builtin_amdgcn_wmma_f32_16x16x4_f32(
                false, afrag[kk], false, bf0, (short)0, acc0, false, false);
            acc1 = __builtin_amdgcn_wmma_f32_16x16x4_f32(
                false, afrag[kk], false, bf1, (short)0, acc1, false, false);
        }
        // Column col = n*16 + lh for both halves; VGPR i -> row m = i + 8*halfsel
        const float bias0 = r[b * DIM + c0];
        const float bias1 = r[b * DIM + c1];
        const float wf0   = Wf[c0];
        const float wf1   = Wf[c1];
        #pragma unroll
        for (int i = 0; i < 8; ++i) {
            logacc[i] += tanhf(acc0[i] + bias0) * wf0;
            logacc[i] += tanhf(acc1[i] + bias1) * wf1;
        }
    }

    // Reduce over the 16 columns held across each 16-lane half.
    #pragma unroll
    for (int i = 0; i < 8; ++i) {
        float s = logacc[i];
        s += __shfl_xor(s, 8, 32);
        s += __shfl_xor(s, 4, 32);
        s += __shfl_xor(s, 2, 32);
        s += __shfl_xor(s, 1, 32);
        logacc[i] = s;
    }
    if (lh == 0) {
        const float bf0 = bfp[0];
        #pragma unroll
        for (int i = 0; i < 8; ++i) {
            const int m = i + halfsel * 8;
            logits[(size_t)b * SKL + s0 + m] = logacc[i] + bf0;
        }
    }
}

// ---------------------------------------------------------------------------
// Kernel 3: per-batch softmax stats (max, sum of exp). One block per batch.
// ---------------------------------------------------------------------------
__global__ __launch_bounds__(256) void softmax_stats_kernel(
    const float* __restrict__ logits,  // (B,SK)
    float* __restrict__ mx,            // (B)
    float* __restrict__ sm)            // (B)
{
    __shared__ float red[256];
    const int b = blockIdx.x;
    const int tid = threadIdx.x;
    const float* lg = logits + (size_t)b * SKL;

    float lmax = -INFINITY;
    for (int s = tid; s < SKL; s += 256) lmax = fmaxf(lmax, lg[s]);
    red[tid] = lmax;
    __syncthreads();
    for (int off = 128; off > 0; off >>= 1) {
        if (tid < off) red[tid] = fmaxf(red[tid], red[tid + off]);
        __syncthreads();
    }
    const float m = red[0];
    __syncthreads();

    float lsum = 0.f;
    for (int s = tid; s < SKL; s += 256) lsum += expf(lg[s] - m);
    red[tid] = lsum;
    __syncthreads();
    for (int off = 128; off > 0; off >>= 1) {
        if (tid < off) red[tid] += red[tid + off];
        __syncthreads();
    }
    if (tid == 0) { mx[b] = m; sm[b] = red[0]; }
}

// ---------------------------------------------------------------------------
// Kernel 4: write attn_sm and per-chunk partial sum of p * v. Streams v once.
// Grid: (SK/CHUNK, B), 256 threads = 2 groups of 128 (d = tid&127).
// ---------------------------------------------------------------------------
#define CHUNK 1024
__global__ __launch_bounds__(256) void attend_kernel(
    const float* __restrict__ logits,  // (B,SK)
    const float* __restrict__ v,       // (B,SK,DIM)
    const float* __restrict__ mx,
    const float* __restrict__ sm,
    float* __restrict__ out_sm,        // (B,SK)  -> d_out + NB*DIM
    float* __restrict__ partial)       // (B, SK/CHUNK, DIM)
{
    __shared__ float sacc[256];
    const int b   = blockIdx.y;
    const int c   = blockIdx.x;
    const int tid = threadIdx.x;
    const int d   = tid & 127;
    const int g   = tid >> 7;
    const int s0  = c * CHUNK;

    const float m   = mx[b];
    const float inv = 1.f / sm[b];
    const float* lg = logits + (size_t)b * SKL;
    const float* vb = v + (size_t)b * SKL * DIM;

    float acc = 0.f;
    for (int s = s0 + g; s < s0 + CHUNK; s += 2) {
        const float p = expf(lg[s] - m) * inv;     // broadcast load across 128 lanes
        acc += p * vb[(size_t)s * DIM + d];
    }
    // attn_sm output for this chunk
    for (int s = s0 + tid; s < s0 + CHUNK; s += 256)
        out_sm[(size_t)b * SKL + s] = expf(lg[s] - m) * inv;

    sacc[tid] = acc;
    __syncthreads();
    if (g == 0)
        partial[((size_t)b * (SKL / CHUNK) + c) * DIM + d] = sacc[d] + sacc[128 + d];
}

// ---------------------------------------------------------------------------
// Kernel 5: deterministic chunk reduction -> attn (B,DIM)
// ---------------------------------------------------------------------------
__global__ __launch_bounds__(128) void reduce_attn_kernel(
    const float* __restrict__ partial,  // (B, SK/CHUNK, DIM)
    float* __restrict__ attn)           // (B,DIM) = d_out
{
    const int b = blockIdx.x;
    const int d = threadIdx.x;
    float acc = 0.f;
    #pragma unroll
    for (int c = 0; c < SKL / CHUNK; ++c)
        acc += partial[((size_t)b * (SKL / CHUNK) + c) * DIM + d];
    attn[b * DIM + d] = acc;
}

// ---------------------------------------------------------------------------
extern "C" void kernel_launch(void* const* d_in, const int* in_sizes, int n_in,
                              void* d_out, int out_size, void* d_ws, size_t ws_size,
                              hipStream_t stream) {
    const float* q  = (const float*)d_in[0];   // (32,1,128)
    const float* k  = (const float*)d_in[1];   // (32,8192,128)
    const float* v  = (const float*)d_in[2];   // (32,8192,128)
    const float* W1 = (const float*)d_in[3];   // (128,128)
    const float* W2 = (const float*)d_in[4];   // (128,128)
    const float* Wf = (const float*)d_in[5];   // (128,1)
    const float* bf = (const float*)d_in[6];   // (1,)

    float* out      = (float*)d_out;
    float* attn     = out;                 // (B,DIM)   = 4096 floats
    float* attn_sm  = out + NB * DIM;      // (B,SK,1)  = 262144 floats

    // Workspace layout
    char*  ws       = (char*)d_ws;
    float* r        = (float*)ws;                                   ws += (size_t)NB * DIM * sizeof(float);
    float* logits   = (float*)ws;                                   ws += (size_t)NB * SKL * sizeof(float);
    float* mx       = (float*)ws;                                   ws += (size_t)NB * sizeof(float);
    float* sm       = (float*)ws;                                   ws += (size_t)NB * sizeof(float);
    float* partial  = (float*)ws;  // (B, SK/CHUNK, DIM)

    qproj_kernel<<<NB, 128, 0, stream>>>(q, W2, r);
    logits_kernel<<<dim3(SKL / 128, NB), 256, 0, stream>>>(k, W1, Wf, bf, r, logits);
    softmax_stats_kernel<<<NB, 256, 0, stream>>>(logits, mx, sm);
    attend_kernel<<<dim3(SKL / CHUNK, NB), 256, 0, stream>>>(logits, v, mx, sm, attn_sm, partial);
    reduce_attn_kernel<<<NB, 128, 0, stream>>>(partial, attn);
}